// TargetAttention_17712445129418
// MI455X (gfx1250) — compile-verified
//
#include <hip/hip_runtime.h>
#include <hip/hip_bf16.h>
#include <math.h>

typedef float v2f __attribute__((ext_vector_type(2)));
typedef float v4f __attribute__((ext_vector_type(4)));
typedef float v8f __attribute__((ext_vector_type(8)));

#define DIM 64

// ---------------------------------------------------------------------------
// One wave computes a 16x16 f32 tile of  C = A(16xK=64) * B(64x16)  via 16
// chained V_WMMA_F32_16X16X4_F32 ops (wave32).
//   A element (m,k):  AT ? A[k*64 + m0+m] : A[(m0+m)*64 + k]
//   B element (k,n):  BT ? B[(n0+n)*64 + k] : B[k*64 + n0+n]
// Fragment layout per CDNA5 ISA 7.12.2 (32-bit A 16x4 / B, C/D 16x16):
//   lanes 0-15 carry M(or N)=lane; lanes 16-31 shift K by +2 (A/B), M by +8 (C/D).
// Contiguous (row-major K-adjacent) fragment halves load as one b64.
// ---------------------------------------------------------------------------
template <bool AT, bool BT>
__device__ __forceinline__ v8f gemm_tile_k64(const float* __restrict__ A,
                                             const float* __restrict__ Bm,
                                             int m0, int n0) {
  const int lane = threadIdx.x & 31;
  const int half = lane >> 4;   // 0: lanes 0-15, 1: lanes 16-31
  const int lm   = lane & 15;
  v8f c = {};
#pragma unroll
  for (int k4 = 0; k4 < DIM; k4 += 4) {
    const int kk = k4 + 2 * half;        // VGPR0 holds K=kk, VGPR1 holds K=kk+1
    v2f a, b;
    if (AT) {            // A column-major read: stride DIM between K's
      a.x = A[(size_t)(kk    ) * DIM + m0 + lm];
      a.y = A[(size_t)(kk + 1) * DIM + m0 + lm];
    } else {             // contiguous pair, 8B aligned -> single b64
      a = *(const v2f*)(A + (size_t)(m0 + lm) * DIM + kk);
    }
    if (BT) {            // B transposed read: contiguous pair -> single b64
      b = *(const v2f*)(Bm + (size_t)(n0 + lm) * DIM + kk);
    } else {
      b.x = Bm[(size_t)(kk    ) * DIM + n0 + lm];
      b.y = Bm[(size_t)(kk + 1) * DIM + n0 + lm];
    }
    c = __builtin_amdgcn_wmma_f32_16x16x4_f32(
        /*neg_a=*/false, a, /*neg_b=*/false, b,
        /*c_mod=*/(short)0, c, /*reuse_a=*/false, /*reuse_b=*/false);
  }
  return c;
}

__device__ __forceinline__ void store_tile16x16(float* __restrict__ D, int ld,
                                                int m0, int n0, v8f c) {
  const int lane = threadIdx.x & 31;
  const int half = lane >> 4;
  const int lm   = lane & 15;
#pragma unroll
  for (int r = 0; r < 8; ++r)
    D[(size_t)(m0 + r + 8 * half) * ld + n0 + lm] = c[r];
}

// ---------------------------------------------------------------------------
// Kernel A: M = W_target^T @ W_item  (64x64).  One block, 16 waves, each wave
// owns one 16x16 tile.  M[d,d'] = sum_e Wt[e,d] * Wi[e,d'].
// ---------------------------------------------------------------------------
__global__ __launch_bounds__(512) void make_M_kernel(const float* __restrict__ Wt,
                                                     const float* __restrict__ Wi,
                                                     float* __restrict__ M) {
  const int wid = threadIdx.x >> 5;          // 0..15
  const int mt = (wid >> 2) * 16;
  const int nt = (wid & 3) * 16;
  v8f c = gemm_tile_k64<true, false>(Wt, Wi, mt, nt);   // A = Wt^T, B = Wi
  store_tile16x16(M, DIM, mt, nt, c);
}

// ---------------------------------------------------------------------------
// Kernel B: qeff = target_key @ M   ([B,64] x [64,64]).
// Grid: B/16 blocks, 4 waves/block; wave w owns N-tile w.
// ---------------------------------------------------------------------------
__global__ __launch_bounds__(128) void proj_q_kernel(const float* __restrict__ tk,
                                                     const float* __restrict__ M,
                                                     float* __restrict__ qc) {
  const int nt = (threadIdx.x >> 5) * 16;
  const int b0 = blockIdx.x * 16;
  v8f c = gemm_tile_k64<false, false>(tk, M, b0, nt);
  store_tile16x16(qc, DIM, b0, nt, c);
}

// ---------------------------------------------------------------------------
// Kernel C: per batch row b — scores over raw item_keys, masked softmax over L,
// weighted reduction over raw item_values.  qc holds qeff on entry and ctx on
// exit (block reads-then-writes only its own row: safe + deterministic).
// Memory-bound pass: 420 MB streamed exactly once, > 2x the 192 MB L2, so all
// bulk loads are non-temporal (TH=NT) to avoid sweeping the cache.
// ---------------------------------------------------------------------------
__global__ __launch_bounds__(256) void attn_core_kernel(float* __restrict__ qc,
                                                        const float* __restrict__ Kd,
                                                        const float* __restrict__ Vd,
                                                        const int* __restrict__ mask,
                                                        int L) {
  __shared__ float s_q[DIM];
  __shared__ float s_w[256];
  __shared__ float s_red[256];
  __shared__ float s_part[256];
  const int t = threadIdx.x;
  const int b = blockIdx.x;

  if (t < DIM) s_q[t] = qc[(size_t)b * DIM + t];
  __syncthreads();

  // scores[l] = (qeff[b] . item_keys[b,l]) / 8 + mask bias
  float sc = -3.0e38f;
  if (t < L) {
    const v4f* row = (const v4f*)(Kd + ((size_t)b * L + t) * DIM);
    float acc = 0.f;
#pragma unroll
    for (int i = 0; i < DIM / 4; ++i) {
      v4f x = __builtin_nontemporal_load(row + i);
      acc += x.x * s_q[4 * i] + x.y * s_q[4 * i + 1] +
             x.z * s_q[4 * i + 2] + x.w * s_q[4 * i + 3];
    }
    sc = acc * 0.125f + (mask[(size_t)b * L + t] ? 0.f : -1.0e8f);
  }
  s_red[t] = sc;
  __syncthreads();
  for (int s = 128; s > 0; s >>= 1) {               // max reduce
    if (t < s) s_red[t] = fmaxf(s_red[t], s_red[t + s]);
    __syncthreads();
  }
  const float mx = s_red[0];
  __syncthreads();

  const float p = (t < L) ? __expf(sc - mx) : 0.f;
  s_w[t]   = p;
  s_red[t] = p;
  __syncthreads();
  for (int s = 128; s > 0; s >>= 1) {               // sum reduce
    if (t < s) s_red[t] += s_red[t + s];
    __syncthreads();
  }
  const float inv = 1.0f / s_red[0];

  // ctx[d] = inv * sum_l p[l] * item_values[b,l,d]; coalesced over d.
  const int d = t & (DIM - 1);
  const int g = t >> 6;                             // 4 l-groups
  float acc = 0.f;
  for (int l = g; l < L; l += 4)
    acc += s_w[l] * __builtin_nontemporal_load(Vd + ((size_t)b * L + l) * DIM + d);
  s_part[t] = acc;
  __syncthreads();
  if (g == 0)
    qc[(size_t)b * DIM + d] =
        (s_part[d] + s_part[DIM + d] + s_part[2 * DIM + d] + s_part[3 * DIM + d]) * inv;
}

// ---------------------------------------------------------------------------
// Kernel D: out = ctx @ W_value^T   ([B,64] x [64,64], B transposed-read).
// ---------------------------------------------------------------------------
__global__ __launch_bounds__(128) void proj_out_kernel(const float* __restrict__ ctx,
                                                       const float* __restrict__ Wv,
                                                       float* __restrict__ out) {
  const int nt = (threadIdx.x >> 5) * 16;
  const int b0 = blockIdx.x * 16;
  v8f c = gemm_tile_k64<false, true>(ctx, Wv, b0, nt);  // B(k=d, n=e) = Wv[e*64+d]
  store_tile16x16(out, DIM, b0, nt, c);
}

extern "C" void kernel_launch(void* const* d_in, const int* in_sizes, int n_in,
                              void* d_out, int out_size, void* d_ws, size_t ws_size,
                              hipStream_t stream) {
  const float* tk = (const float*)d_in[0];   // target_key  [B,64]
  const float* ik = (const float*)d_in[1];   // item_keys   [B,L,64]
  const float* iv = (const float*)d_in[2];   // item_values [B,L,64]
  const int*   mk = (const int*)d_in[3];     // mask        [B,L]
  const float* Wt = (const float*)d_in[4];   // W_target    [64,64]
  const float* Wi = (const float*)d_in[5];   // W_item      [64,64]
  const float* Wv = (const float*)d_in[6];   // W_value     [64,64]
  float* out = (float*)d_out;

  const int B = in_sizes[0] / DIM;           // 4096
  const int L = in_sizes[3] / B;             // 200

  float* Mw = (float*)d_ws;                  // 64*64 floats
  float* qc = Mw + DIM * DIM;                // B*64 floats (qeff, then ctx)

  make_M_kernel<<<1, 512, 0, stream>>>(Wt, Wi, Mw);
  proj_q_kernel<<<B / 16, 128, 0, stream>>>(tk, Mw, qc);
  attn_core_kernel<<<B, 256, 0, stream>>>(qc, ik, iv, mk, L);
  proj_out_kernel<<<B / 16, 128, 0, stream>>>(qc, Wv, out);
}